// SuperGNNModel_68281390072314
// MI455X (gfx1250) — compile-verified
//
#include <hip/hip_runtime.h>
#include <math.h>

typedef __attribute__((ext_vector_type(2))) float v2f;
typedef __attribute__((ext_vector_type(8))) float v8f;

#define ORD_NEG_INF ((int)0x807FFFFF)  // f2ord(-inf)

__device__ __forceinline__ int f2ord(float f) {
    int i = __float_as_int(f);
    return i ^ ((i >> 31) & 0x7fffffff);
}
__device__ __forceinline__ float ord2f(int i) {
    return __int_as_float(i ^ ((i >> 31) & 0x7fffffff));
}

// ---------------- fills ----------------
__global__ void k_fill_f32(float* __restrict__ p, size_t n, float v) {
    size_t i = (size_t)blockIdx.x * blockDim.x + threadIdx.x;
    size_t s = (size_t)gridDim.x * blockDim.x;
    for (; i < n; i += s) p[i] = v;
}
__global__ void k_fill_i32(int* __restrict__ p, size_t n, int v) {
    size_t i = (size_t)blockIdx.x * blockDim.x + threadIdx.x;
    size_t s = (size_t)gridDim.x * blockDim.x;
    for (; i < n; i += s) p[i] = v;
}

// ---------------- self-loop mean edge_attr ----------------
__global__ void k_deg(const int* __restrict__ dst, const float* __restrict__ eattr,
                      float* __restrict__ deg, float* __restrict__ msum, int E) {
    int e = blockIdx.x * blockDim.x + threadIdx.x;
    if (e >= E) return;
    int d = dst[e];
    atomicAdd(&deg[d], 1.0f);
    atomicAdd(&msum[(size_t)d * 3 + 0], eattr[(size_t)e * 3 + 0]);
    atomicAdd(&msum[(size_t)d * 3 + 1], eattr[(size_t)e * 3 + 1]);
    atomicAdd(&msum[(size_t)d * 3 + 2], eattr[(size_t)e * 3 + 2]);
}
__global__ void k_mean(float* __restrict__ msum, const float* __restrict__ deg, int N) {
    int n = blockIdx.x * blockDim.x + threadIdx.x;
    if (n >= N) return;
    float dv = fmaxf(deg[n], 1.0f);
    msum[(size_t)n * 3 + 0] /= dv;
    msum[(size_t)n * 3 + 1] /= dv;
    msum[(size_t)n * 3 + 2] /= dv;
}

// ---------------- WMMA f32 GEMM: C[M,Nc] = A[M,K] @ B[K,Nc] ----------------
// One wave computes a 16 x (16*TW) strip via V_WMMA_F32_16X16X4_F32, K stepped
// by 4. The A fragment (one b64 load per k-step) is reused across TW WMMAs.
// A 16x4 layout : lane<16 -> M=lane, K=k+{0,1};  lane>=16 -> M=lane-16, K=k+{2,3}
// B 4x16 layout : mirrored (row striped across lanes)
// C 16x16 layout: VGPR r -> M = r + (lane>=16 ? 8:0), N = lane&15
template <int TW>
__global__ void k_gemm(const float* __restrict__ A, const float* __restrict__ B,
                       float* __restrict__ C, int tilesM, int K, int Nc) {
    int wave = (int)(((size_t)blockIdx.x * blockDim.x + threadIdx.x) >> 5);
    int lane = threadIdx.x & 31;
    int strips = (Nc >> 4) / TW;
    if (wave >= tilesM * strips) return;
    int tm = wave / strips, ts = wave % strips;
    int row  = tm * 16 + (lane & 15);
    int col0 = ts * (16 * TW) + (lane & 15);
    int koff = (lane >> 4) << 1;  // 0 or 2

#if __has_builtin(__builtin_amdgcn_wmma_f32_16x16x4_f32)
    v8f acc[TW];
#pragma unroll
    for (int t = 0; t < TW; ++t) acc[t] = (v8f){};
    for (int k = 0; k < K; k += 4) {
        // A fragment: two consecutive floats -> single 8B-aligned b64 load
        v2f a = *(const v2f*)(A + (size_t)row * K + (k + koff));
        const float* Brow0 = B + (size_t)(k + koff) * Nc;
        const float* Brow1 = B + (size_t)(k + koff + 1) * Nc;
#pragma unroll
        for (int t = 0; t < TW; ++t) {
            v2f b;
            b.x = Brow0[col0 + 16 * t];
            b.y = Brow1[col0 + 16 * t];
            acc[t] = __builtin_amdgcn_wmma_f32_16x16x4_f32(
                /*neg_a=*/false, a, /*neg_b=*/false, b,
                /*c_mod=*/(short)0, acc[t], /*reuse_a=*/false, /*reuse_b=*/false);
        }
    }
    int rbase = tm * 16 + ((lane >> 4) << 3);
#pragma unroll
    for (int t = 0; t < TW; ++t)
#pragma unroll
        for (int r = 0; r < 8; ++r)
            C[(size_t)(rbase + r) * Nc + col0 + 16 * t] = acc[t][r];
#else
    // scalar fallback (should not be needed on gfx1250)
    int rbase = tm * 16 + ((lane >> 4) << 3);
    for (int t = 0; t < TW; ++t) {
        float acc[8] = {};
        for (int k = 0; k < K; ++k)
#pragma unroll
            for (int r = 0; r < 8; ++r)
                acc[r] += A[(size_t)(rbase + r) * K + k] * B[(size_t)k * Nc + col0 + 16 * t];
#pragma unroll
        for (int r = 0; r < 8; ++r)
            C[(size_t)(rbase + r) * Nc + col0 + 16 * t] = acc[r];
    }
#endif
}

// ---------------- per-node attention scores ----------------
__global__ void k_nscore(const float* __restrict__ xp, const float* __restrict__ asrc,
                         const float* __restrict__ adst, float* __restrict__ ssrc,
                         float* __restrict__ sdst, int N, int H, int O) {
    int i = blockIdx.x * blockDim.x + threadIdx.x;
    if (i >= N * H) return;
    int n = i / H, h = i % H;
    const float* xr = xp + (size_t)n * H * O + (size_t)h * O;
    float s1 = 0.f, s2 = 0.f;
    for (int o = 0; o < O; ++o) {
        float v = xr[o];
        s1 += v * asrc[h * O + o];
        s2 += v * adst[h * O + o];
    }
    ssrc[i] = s1;
    sdst[i] = s2;
}

// ce[d*H+h] = sum_o We[d, h*O+o] * a_e[h,o]   (folds edge projection into 3xH)
__global__ void k_ce(const float* __restrict__ We, const float* __restrict__ ae,
                     float* __restrict__ ce, int H, int O) {
    int i = threadIdx.x;
    if (i >= 3 * H) return;
    int d = i / H, h = i % H;
    float s = 0.f;
    for (int o = 0; o < O; ++o)
        s += We[(size_t)d * H * O + h * O + o] * ae[h * O + o];
    ce[i] = s;
}

// ---------------- edge attention logits + segment max ----------------
__global__ void k_alpha(const int* __restrict__ src, const int* __restrict__ dst,
                        const float* __restrict__ eattr, const float* __restrict__ mattr,
                        const float* __restrict__ ssrc, const float* __restrict__ sdst,
                        const float* __restrict__ ce, float* __restrict__ alpha,
                        int* __restrict__ mmax, int E, int N, int H) {
    int i = blockIdx.x * blockDim.x + threadIdx.x;
    int total = (E + N) * H;
    if (i >= total) return;
    int e = i / H, h = i % H;
    int s, d;
    const float* a3;
    if (e < E) { s = src[e]; d = dst[e]; a3 = eattr + (size_t)e * 3; }
    else       { s = d = e - E;          a3 = mattr + (size_t)(e - E) * 3; }
    float a = ssrc[s * H + h] + sdst[d * H + h]
            + a3[0] * ce[0 * H + h] + a3[1] * ce[1 * H + h] + a3[2] * ce[2 * H + h];
    a = a > 0.f ? a : 0.2f * a;  // LeakyReLU(0.2)
    alpha[i] = a;
    atomicMax(&mmax[d * H + h], f2ord(a));
}

// ---------------- exp(alpha - max) + segment sum ----------------
__global__ void k_exp(const int* __restrict__ dst, float* __restrict__ alpha,
                      const int* __restrict__ mmax, float* __restrict__ denom,
                      int E, int N, int H) {
    int i = blockIdx.x * blockDim.x + threadIdx.x;
    int total = (E + N) * H;
    if (i >= total) return;
    int e = i / H, h = i % H;
    int d = (e < E) ? dst[e] : (e - E);
    float ex = expf(alpha[i] - ord2f(mmax[d * H + h]));
    alpha[i] = ex;
    atomicAdd(&denom[d * H + h], ex);
}

// ---------------- scatter aggregation: one wave per edge ----------------
__global__ void k_scatter(const int* __restrict__ src, const int* __restrict__ dst,
                          const float* __restrict__ xp, const float* __restrict__ ea,
                          float* __restrict__ agg, int E, int N, int H, int O) {
    int wave = (int)(((size_t)blockIdx.x * blockDim.x + threadIdx.x) >> 5);
    int lane = threadIdx.x & 31;
    if (wave >= E + N) return;
    int s = (wave < E) ? src[wave] : wave - E;
    int d = (wave < E) ? dst[wave] : wave - E;
    int HO = H * O;
    const float* xrow = xp + (size_t)s * HO;
    float* arow = agg + (size_t)d * HO;
    // start pulling the gathered source row while edge weights resolve
    __builtin_prefetch(xrow + lane, 0, 0);
    for (int c = lane; c < HO; c += 32) {
        float w = ea[(size_t)wave * H + c / O];
        atomicAdd(&arow[c], w * xrow[c]);
    }
}

// ---------------- normalize + bias + (optional) ELU ----------------
__global__ void k_final(float* __restrict__ agg, const float* __restrict__ denom,
                        const float* __restrict__ b, int N, int H, int O, int do_elu) {
    int i = blockIdx.x * blockDim.x + threadIdx.x;
    int HO = H * O;
    if (i >= N * HO) return;
    int n = i / HO, c = i % HO, h = c / O;
    float v = agg[i] / denom[n * H + h] + b[c];
    if (do_elu) v = v > 0.f ? v : expf(v) - 1.f;
    agg[i] = v;
}

static inline int cdiv_i(long a, long b) { return (int)((a + b - 1) / b); }

extern "C" void kernel_launch(void* const* d_in, const int* in_sizes, int n_in,
                              void* d_out, int out_size, void* d_ws, size_t ws_size,
                              hipStream_t stream) {
    const float* x     = (const float*)d_in[0];
    const int*   eidx  = (const int*)d_in[1];
    const float* eattr = (const float*)d_in[2];
    const float* W[3]    = {(const float*)d_in[3],  (const float*)d_in[9],  (const float*)d_in[15]};
    const float* asrc[3] = {(const float*)d_in[4],  (const float*)d_in[10], (const float*)d_in[16]};
    const float* adst[3] = {(const float*)d_in[5],  (const float*)d_in[11], (const float*)d_in[17]};
    const float* Wep[3]  = {(const float*)d_in[6],  (const float*)d_in[12], (const float*)d_in[18]};
    const float* aep[3]  = {(const float*)d_in[7],  (const float*)d_in[13], (const float*)d_in[19]};
    const float* bia[3]  = {(const float*)d_in[8],  (const float*)d_in[14], (const float*)d_in[20]};

    const int N = in_sizes[0] / 128;
    const int E = in_sizes[1] / 2;
    const int* src = eidx;
    const int* dst = eidx + E;

    // workspace carve-up (floats)
    float* w = (float*)d_ws;
    size_t o = 0;
    float* deg   = w + o; o += (size_t)N;
    float* mattr = w + o; o += (size_t)3 * N;       // contiguous with deg
    float* xp    = w + o; o += (size_t)N * 256;
    float* hA    = w + o; o += (size_t)N * 256;
    float* hB    = w + o; o += (size_t)N * 64;
    float* ssrc  = w + o; o += (size_t)N * 4;
    float* sdst  = w + o; o += (size_t)N * 4;
    int*   mmax  = (int*)(w + o); o += (size_t)N * 4;
    float* denom = w + o; o += (size_t)N * 4;
    float* eaw   = w + o; o += (size_t)(E + N) * 4;
    float* ce    = w + o; o += 16;

    // ---- self-loop mean edge_attr ----
    k_fill_f32<<<2048, 256, 0, stream>>>(deg, (size_t)4 * N, 0.f);  // deg + mattr
    k_deg<<<cdiv_i(E, 256), 256, 0, stream>>>(dst, eattr, deg, mattr, E);
    k_mean<<<cdiv_i(N, 256), 256, 0, stream>>>(mattr, deg, N);

    struct L { int K, H, O; };
    const L Ls[3] = {{128, 4, 64}, {256, 2, 32}, {64, 1, 16}};
    const float* hin = x;
    float* aggs[3] = {hA, hB, (float*)d_out};

    for (int l = 0; l < 3; ++l) {
        const int K = Ls[l].K, H = Ls[l].H, O = Ls[l].O, HO = H * O;
        float* agg = aggs[l];
        const int tilesM = N / 16, tilesN = HO / 16;

        if (tilesN % 4 == 0) {
            long waves = (long)tilesM * (tilesN / 4);
            k_gemm<4><<<cdiv_i(waves * 32, 256), 256, 0, stream>>>(hin, W[l], xp, tilesM, K, HO);
        } else {
            long waves = (long)tilesM * tilesN;
            k_gemm<1><<<cdiv_i(waves * 32, 256), 256, 0, stream>>>(hin, W[l], xp, tilesM, K, HO);
        }
        k_nscore<<<cdiv_i((long)N * H, 256), 256, 0, stream>>>(xp, asrc[l], adst[l], ssrc, sdst, N, H, O);
        k_ce<<<1, 64, 0, stream>>>(Wep[l], aep[l], ce, H, O);
        k_fill_i32<<<cdiv_i((long)N * H, 256), 256, 0, stream>>>(mmax, (size_t)N * H, ORD_NEG_INF);
        k_fill_f32<<<cdiv_i((long)N * H, 256), 256, 0, stream>>>(denom, (size_t)N * H, 0.f);
        k_alpha<<<cdiv_i((long)(E + N) * H, 256), 256, 0, stream>>>(src, dst, eattr, mattr, ssrc, sdst,
                                                                    ce, eaw, mmax, E, N, H);
        k_exp<<<cdiv_i((long)(E + N) * H, 256), 256, 0, stream>>>(dst, eaw, mmax, denom, E, N, H);
        k_fill_f32<<<2048, 256, 0, stream>>>(agg, (size_t)N * HO, 0.f);
        k_scatter<<<cdiv_i((long)(E + N), 8), 256, 0, stream>>>(src, dst, xp, eaw, agg, E, N, H, O);
        k_final<<<cdiv_i((long)N * HO, 256), 256, 0, stream>>>(agg, denom, bia[l], N, H, O, (l < 2) ? 1 : 0);

        hin = agg;
    }
}